// PDFSampler_54099408060632
// MI455X (gfx1250) — compile-verified
//
#include <hip/hip_runtime.h>
#include <stdint.h>

// PDF resample (NeRF importance sampling) for MI455X / gfx1250.
// Memory-bound op (~671 MB @ 23.3 TB/s ~= 29us floor). Strategy:
//  - async global->LDS staging (ASYNCcnt path, global_load_async_to_lds_b128)
//  - exclusive prefix scan of pdf via chained V_WMMA_F32_16X16X4_F32
//    (cdf = min(1, pdf @ strict-upper-triangular-ones), 16 rays/wave)
//  - inverse-CDF sampling + merge-by-rank (no full sort), all in LDS
//  - one store per output element

#define NUM_BINS       64
#define NUM_SAMPLES    128
#define OUT_PER_RAY    192
#define RAYS_PER_BLOCK 32
#define BLOCK_THREADS  64      // 2 waves (wave32), 16 rays per wave for WMMA
#define EPS_F          1e-5f

// LDS row strides (floats). CDF stride 68 -> 272B rows: 16B aligned for B128
// async copies, and bank shift of 4 makes the column-uniform A-tile reads
// (lanes 0..15 read 16 different rows at the same column) conflict-free.
#define CDF_STRIDE 68
#define TS_STRIDE  64
#define NT_STRIDE  129

typedef __attribute__((ext_vector_type(2))) float v2f;
typedef __attribute__((ext_vector_type(8))) float v8f;
typedef __attribute__((ext_vector_type(4))) int   v4i;

#if __has_builtin(__builtin_amdgcn_global_load_async_to_lds_b128)
#define HAVE_ASYNC_LDS 1
typedef __attribute__((address_space(1))) v4i g1_v4i;
typedef __attribute__((address_space(3))) v4i l3_v4i;
// Flat LDS pointers hold the LDS byte offset in addr[31:0] (ISA aperture
// rules), so integer truncation yields the correct AS(3) pointer. Global
// flat pointers are numerically identical to AS(1) pointers.
static __device__ __forceinline__ l3_v4i* lds_cast(const void* p) {
  return (l3_v4i*)(uint32_t)(uintptr_t)p;
}
static __device__ __forceinline__ g1_v4i* g1_cast(const void* p) {
  return (g1_v4i*)(uintptr_t)p;
}
#endif

__launch_bounds__(BLOCK_THREADS)
__global__ void pdf_resample_wmma(const float* __restrict__ weights,
                                  const float* __restrict__ tsin,
                                  float* __restrict__ out)
{
  __shared__ __align__(16) float s_cdf[RAYS_PER_BLOCK * CDF_STRIDE]; // w -> pdf -> cdf
  __shared__ __align__(16) float s_ts [RAYS_PER_BLOCK * TS_STRIDE];
  __shared__ float s_nt   [RAYS_PER_BLOCK * NT_STRIDE];              // new_ts
  __shared__ float s_carry[RAYS_PER_BLOCK * 4];                      // per-tile scan carry

  const int t = threadIdx.x;
  const long long R0 = (long long)blockIdx.x * RAYS_PER_BLOCK;

  // ---------------- Phase 1: stage weights & ts rows into LDS ----------------
#if defined(HAVE_ASYNC_LDS)
#pragma unroll
  for (int it = 0; it < 8; ++it) {
    const int idx = it * BLOCK_THREADS + t;   // 0..511 -> (row, 16B segment)
    const int row = idx >> 4;
    const int seg = idx & 15;
    __builtin_amdgcn_global_load_async_to_lds_b128(
        g1_cast(weights + (R0 + row) * NUM_BINS + seg * 4),
        lds_cast(&s_cdf[row * CDF_STRIDE + seg * 4]), 0, 0);
    __builtin_amdgcn_global_load_async_to_lds_b128(
        g1_cast(tsin + (R0 + row) * NUM_BINS + seg * 4),
        lds_cast(&s_ts[row * TS_STRIDE + seg * 4]), 0, 0);
  }
#if __has_builtin(__builtin_amdgcn_s_wait_asynccnt)
  __builtin_amdgcn_s_wait_asynccnt(0);
#else
  asm volatile("s_wait_asynccnt 0" ::: "memory");
#endif
#else
#pragma unroll
  for (int it = 0; it < 8; ++it) {
    const int idx = it * BLOCK_THREADS + t;
    const int row = idx >> 4;
    const int seg = idx & 15;
    const float4 w4 = *(const float4*)(weights + (R0 + row) * NUM_BINS + seg * 4);
    const float4 t4 = *(const float4*)(tsin    + (R0 + row) * NUM_BINS + seg * 4);
    *(float4*)&s_cdf[row * CDF_STRIDE + seg * 4] = w4;
    *(float4*)&s_ts [row * TS_STRIDE  + seg * 4] = t4;
  }
#endif
  __syncthreads();

  // ------- Phase 2: per-ray normalization (w -> pdf in place) + carries -------
  if (t < RAYS_PER_BLOCK) {
    float* wrow = &s_cdf[t * CDF_STRIDE];
    float wsum = 0.0f, p16 = 0.0f, p32 = 0.0f, p48 = 0.0f;
    for (int k = 0; k < 63; ++k) {           // w = weights[..., :-1]
      if (k == 16) p16 = wsum;
      if (k == 32) p32 = wsum;
      if (k == 48) p48 = wsum;
      wsum += wrow[k];
    }
    const float padding = fmaxf(0.0f, EPS_F - wsum);
    const float bias  = padding * (1.0f / 63.0f);
    const float scale = 1.0f / (wsum + padding);
    for (int k = 0; k < 63; ++k) wrow[k] = (wrow[k] + bias) * scale;
    wrow[63] = 0.0f;                         // padded pdf lane (never contributes)
    s_carry[t * 4 + 0] = 0.0f;
    s_carry[t * 4 + 1] = (p16 + 16.0f * bias) * scale;
    s_carry[t * 4 + 2] = (p32 + 32.0f * bias) * scale;
    s_carry[t * 4 + 3] = (p48 + 48.0f * bias) * scale;
  }
  __syncthreads();

  // ---- Phase 3: WMMA exclusive scan: cdf[j] = min(1, sum_{k<j} pdf[k]) ----
  // Per wave: 16 rays. D_tile(c) = sum_{q=0..3} A(16x4 pdf cols) @ B_q + 0,
  // B_q[k',j'] = (4q + k' < j'), then add cross-tile carry, clamp, write back
  // into s_cdf (reusing the pdf buffer; columns are written after being read).
  {
    const int lane = t & 31;
    const int wv   = t >> 5;                 // wave id -> rays [16*wv, 16*wv+16)
    const int mrow = (lane & 15) + (wv << 4);
    const int hi   = lane >> 4;              // lane half
    const int kb   = hi << 1;                // K base per §7.12.2: 0 or 2
    const int ncol = lane & 15;              // output column for C/D

    v2f B[4];
#pragma unroll
    for (int q = 0; q < 4; ++q) {
      B[q].x = ((4 * q + kb + 0) < ncol) ? 1.0f : 0.0f;
      B[q].y = ((4 * q + kb + 1) < ncol) ? 1.0f : 0.0f;
    }

#pragma unroll
    for (int c = 0; c < 4; ++c) {
      v8f acc = {0.f, 0.f, 0.f, 0.f, 0.f, 0.f, 0.f, 0.f};
#pragma unroll
      for (int q = 0; q < 4; ++q) {
        const int col = 16 * c + 4 * q + kb;
        v2f a;
        a.x = s_cdf[mrow * CDF_STRIDE + col + 0];
        a.y = s_cdf[mrow * CDF_STRIDE + col + 1];
        acc = __builtin_amdgcn_wmma_f32_16x16x4_f32(
            false, a, false, B[q], (short)0, acc, false, false);
      }
#pragma unroll
      for (int v = 0; v < 8; ++v) {
        const int ray = (wv << 4) + v + (hi << 3);   // C/D row mapping
        float val = acc[v] + s_carry[ray * 4 + c];
        val = fminf(val, 1.0f);
        s_cdf[ray * CDF_STRIDE + 16 * c + ncol] = val;
      }
    }
  }
  __syncthreads();

  // ------------- Phase 4: inverse-CDF sampling (128 u's per ray) -------------
  {
    const int ray = t & (RAYS_PER_BLOCK - 1);
    const int j0  = (t >> 5) << 6;           // wave0: u[0..63], wave1: u[64..127]
    const float* crow = &s_cdf[ray * CDF_STRIDE];
    const float* trow = &s_ts [ray * TS_STRIDE];
    float* nrow = &s_nt[ray * NT_STRIDE];
    for (int i = 0; i < 64; ++i) {
      const int j = j0 + i;
      const float u = (float)j / 127.0f;     // linspace(0,1,128)
      // searchsorted(cdf, u, side='right'): first idx with cdf[idx] > u
      int lo = 0, len = 64;
      while (len > 0) {
        const int h = len >> 1;
        if (crow[lo + h] <= u) { lo += h + 1; len -= h + 1; } else len = h;
      }
      const int below = (lo > 1) ? lo - 1 : 0;
      const int above = (lo < 63) ? lo : 63;
      const float cb = crow[below], ca = crow[above];
      const float bb = trow[below], ba = trow[above];
      float den = ca - cb;
      den = (den < EPS_F) ? 1.0f : den;
      const float tt = (u - cb) / den;
      nrow[j] = bb + tt * (ba - bb);
    }
  }
  __syncthreads();

  // --------- Phase 5: merge-by-rank of sorted ts(64) and new_ts(128) ---------
  {
    const int ray = t & (RAYS_PER_BLOCK - 1);
    const int e0  = (t >> 5) * 96;           // 96 merged elements per thread
    const float* trow = &s_ts[ray * TS_STRIDE];
    const float* nrow = &s_nt[ray * NT_STRIDE];
    float* __restrict__ orow = out + (R0 + ray) * OUT_PER_RAY;
    for (int i = 0; i < 96; ++i) {
      const int e = e0 + i;
      float val;
      int pos;
      if (e < NUM_BINS) {
        val = trow[e];
        // rank among new_ts: count of new_ts < val (lower_bound)
        int lo = 0, len = NUM_SAMPLES;
        while (len > 0) {
          const int h = len >> 1;
          if (nrow[lo + h] < val) { lo += h + 1; len -= h + 1; } else len = h;
        }
        pos = e + lo;
      } else {
        const int j = e - NUM_BINS;
        val = nrow[j];
        // rank among ts: count of ts <= val (upper_bound)
        int lo = 0, len = NUM_BINS;
        while (len > 0) {
          const int h = len >> 1;
          if (trow[lo + h] <= val) { lo += h + 1; len -= h + 1; } else len = h;
        }
        pos = j + lo;
      }
      orow[pos] = val;                        // bijection onto [0,192)
    }
  }
}

extern "C" void kernel_launch(void* const* d_in, const int* in_sizes, int n_in,
                              void* d_out, int out_size, void* d_ws, size_t ws_size,
                              hipStream_t stream) {
  (void)n_in; (void)out_size; (void)d_ws; (void)ws_size;
  const float* weights = (const float*)d_in[0];
  const float* tsin    = (const float*)d_in[1];
  float* out           = (float*)d_out;
  const int n_rays = in_sizes[0] / NUM_BINS;       // 524288
  const int blocks = n_rays / RAYS_PER_BLOCK;      // 16384
  pdf_resample_wmma<<<blocks, BLOCK_THREADS, 0, stream>>>(weights, tsin, out);
}